// LSTMLayer_75118978007540
// MI455X (gfx1250) — compile-verified
//
#include <hip/hip_runtime.h>
#include <cstddef>

// ---------------------------------------------------------------------------
// LSTM layer for MI455X (gfx1250), f32 end-to-end using V_WMMA_F32_16X16X4_F32.
//   Phase 0: permute weights into per-WG gate-grouped column layout, zero bars
//   Phase 1: Y = X @ Wx + bias  (double-buffered WMMA GEMM, M=65536,K=512,N=2048)
//   Phase 2: persistent 32-WG team, weights resident in LDS, serial t-loop,
//            async global->LDS broadcast of h between steps (ASYNCcnt path)
// ---------------------------------------------------------------------------

typedef float v2f __attribute__((ext_vector_type(2)));
typedef float v8f __attribute__((ext_vector_type(8)));

#define B_   32
#define T_   2048
#define D_   512
#define L_   512
#define N_   2048      // 4*L permuted columns: per 16-j group [i|f|o|mem]
#define NWG  32        // phase-2 team size (one WG per 64-column slice)
#define SW   524       // LDS row stride (floats): even (b64-aligned frags),
                       // 524%4==0 (b128-aligned rows), 524 mod 64 == 12 so the
                       // 16 fragment lanes hit 16 distinct banks, and the +2
                       // hi-half offset provably never collides with lo-half
#define SZ   68        // LDS row stride for Z exchange
#define SA1  28        // phase-1 LDS stride: even, 28 mod 64 bank-clean too

// --------------------------- gfx1250 async helpers -------------------------
__device__ __forceinline__ void async_copy_b128(float* lds_dst, const float* gsrc)
{
    // GLOBAL_LOAD_ASYNC_TO_LDS_B128 (ISA 15.18.3 op 98): VDST = per-lane LDS
    // byte address, VADDR = 64-bit global address. A generic pointer into the
    // LDS aperture carries the LDS byte offset in its low 32 bits (ISA 10.2).
    unsigned       lds = (unsigned)(size_t)lds_dst;
    unsigned long long ga = (unsigned long long)(size_t)gsrc;
    asm volatile("global_load_async_to_lds_b128 %0, %1, off"
                 :: "v"(lds), "v"(ga) : "memory");
}
__device__ __forceinline__ void wait_async0()
{
#if defined(__has_builtin) && __has_builtin(__builtin_amdgcn_s_wait_asynccnt)
    __builtin_amdgcn_s_wait_asynccnt(0);
#else
    asm volatile("s_wait_asynccnt 0x0" ::: "memory");
#endif
}

// ---------------------------------------------------------------------------
// Phase 0: build Wperm[1024][2048], bias_perm[2048]; zero per-step barriers.
// nperm = grp*64 + q*16 + jj : q=0 -> w_w[:,j] (i), q=1 -> w_w[:,512+j] (f),
//                              q=2 -> w_w[:,1024+j] (o), q=3 -> m_w[:,j] (mem)
// ---------------------------------------------------------------------------
__global__ __launch_bounds__(256) void lstm_prep_kernel(
    const float* __restrict__ w_w, const float* __restrict__ w_b,
    const float* __restrict__ m_w, const float* __restrict__ m_b,
    float* __restrict__ Wperm, float* __restrict__ biasp,
    unsigned* __restrict__ Bar)
{
    size_t idx = (size_t)blockIdx.x * 256 + threadIdx.x;
    if (idx >= (size_t)1024 * N_) return;
    int nperm = (int)(idx & (N_ - 1));
    int k     = (int)(idx >> 11);
    int grp = nperm >> 6;
    int q   = (nperm >> 4) & 3;
    int jj  = nperm & 15;
    int j   = grp * 16 + jj;
    float v = (q < 3) ? w_w[(size_t)k * (3 * L_) + q * L_ + j]
                      : m_w[(size_t)k * L_ + j];
    Wperm[idx] = v;
    if (k == 0) biasp[nperm] = (q < 3) ? w_b[q * L_ + j] : m_b[j];
    if (idx < T_) Bar[idx] = 0u;
}

// ---------------------------------------------------------------------------
// Phase 1: Y[R][n] = sum_k x_row(R)[k] * Wperm[k][n] + biasp[n],  R = t*32 + b
// 64x128 tile per WG (256 thr, 8 waves); each wave: 4 N-tiles sharing one A
// fragment. Double-buffered LDS staging, b128 global loads.
// ---------------------------------------------------------------------------
__global__ __launch_bounds__(256) void lstm_xproj_kernel(
    const float* __restrict__ x, const float* __restrict__ Wperm,
    const float* __restrict__ biasp, float* __restrict__ Y)
{
    const int nTile = blockIdx.x;   // 0..15  (128 cols each)
    const int mTile = blockIdx.y;   // 0..1023 (64 rows each)
    const int tid  = threadIdx.x;
    const int lane = tid & 31;
    const int wv   = tid >> 5;
    const int l16  = lane & 15;
    const int hi   = lane >> 4;
    const int mt   = wv & 3;        // M-tile 0..3
    const int nh   = wv >> 2;       // N-quad 0..1 (n-tiles nh*4 .. nh*4+3)

    __shared__ float As[2][64 * SA1];   // A[m][k]
    __shared__ float Bs[2][128 * SA1];  // B transposed: Bs[n][k]

    // staging coordinates
    const int ar = tid >> 2, aq = tid & 3;            // A: row, float4 slot
    const int bc4 = (tid & 31) * 4, brk = tid >> 5;   // B: col*4, k-row (0..7)
    const int Rr = mTile * 64 + ar;
    const int bb = Rr & 31, tt = Rr >> 5;
    const float* xrow = x + ((size_t)bb * T_ + tt) * D_;
    const float* wcol = Wperm + (size_t)nTile * 128 + bc4;

    // preload chunk 0 into buffer 0
    {
        float4 a0 = *(const float4*)&xrow[aq * 4];
        float4 b0 = *(const float4*)&wcol[(size_t)brk * N_];
        float4 b1 = *(const float4*)&wcol[(size_t)(brk + 8) * N_];
        *(float4*)&As[0][ar * SA1 + aq * 4] = a0;
        Bs[0][(bc4 + 0) * SA1 + brk] = b0.x;
        Bs[0][(bc4 + 1) * SA1 + brk] = b0.y;
        Bs[0][(bc4 + 2) * SA1 + brk] = b0.z;
        Bs[0][(bc4 + 3) * SA1 + brk] = b0.w;
        Bs[0][(bc4 + 0) * SA1 + brk + 8] = b1.x;
        Bs[0][(bc4 + 1) * SA1 + brk + 8] = b1.y;
        Bs[0][(bc4 + 2) * SA1 + brk + 8] = b1.z;
        Bs[0][(bc4 + 3) * SA1 + brk + 8] = b1.w;
    }
    __syncthreads();

    v8f acc[4] = {};
    int p = 0;
    for (int k0 = 0; k0 < D_; k0 += 16, p ^= 1) {
        const bool more = (k0 + 16) < D_;
        float4 an = {}, bn0 = {}, bn1 = {};
        if (more) {   // issue next chunk's global loads before the WMMA chain
            an  = *(const float4*)&xrow[k0 + 16 + aq * 4];
            bn0 = *(const float4*)&wcol[(size_t)(k0 + 16 + brk) * N_];
            bn1 = *(const float4*)&wcol[(size_t)(k0 + 16 + brk + 8) * N_];
        }
        #pragma unroll
        for (int kk = 0; kk < 16; kk += 4) {
            int kb = kk + 2 * hi;
            v2f a = *(const v2f*)&As[p][(mt * 16 + l16) * SA1 + kb];
            #pragma unroll
            for (int j = 0; j < 4; ++j) {
                v2f bf = *(const v2f*)&Bs[p][((nh * 4 + j) * 16 + l16) * SA1 + kb];
                acc[j] = __builtin_amdgcn_wmma_f32_16x16x4_f32(
                    false, a, false, bf, (short)0, acc[j], false, false);
            }
        }
        if (more) {
            int pn = p ^ 1;
            *(float4*)&As[pn][ar * SA1 + aq * 4] = an;
            Bs[pn][(bc4 + 0) * SA1 + brk] = bn0.x;
            Bs[pn][(bc4 + 1) * SA1 + brk] = bn0.y;
            Bs[pn][(bc4 + 2) * SA1 + brk] = bn0.z;
            Bs[pn][(bc4 + 3) * SA1 + brk] = bn0.w;
            Bs[pn][(bc4 + 0) * SA1 + brk + 8] = bn1.x;
            Bs[pn][(bc4 + 1) * SA1 + brk + 8] = bn1.y;
            Bs[pn][(bc4 + 2) * SA1 + brk + 8] = bn1.z;
            Bs[pn][(bc4 + 3) * SA1 + brk + 8] = bn1.w;
            __syncthreads();
        }
    }

    // epilogue: + bias, store
    #pragma unroll
    for (int j = 0; j < 4; ++j) {
        int n = nTile * 128 + (nh * 4 + j) * 16 + l16;
        float bias = biasp[n];
        #pragma unroll
        for (int r = 0; r < 8; ++r) {
            int R = mTile * 64 + mt * 16 + r + 8 * hi;
            Y[(size_t)R * N_ + n] = acc[j][r] + bias;
        }
    }
}

// ---------------------------------------------------------------------------
// Phase 2: persistent recurrent team. 32 WGs x 256 threads (8 waves).
// WG wg owns permuted columns [wg*64, wg*64+64) == h-columns [wg*16, wg*16+16)
// LDS: Wl[64][SW] (W slice, transposed), Hs[32][SW], Zs[32][SZ], Cs[32][16]
// ---------------------------------------------------------------------------
__global__ __launch_bounds__(256) void lstm_rec_kernel(
    const float* __restrict__ Wperm, const float* __restrict__ Y,
    const float* __restrict__ h0, const float* __restrict__ c0,
    float* __restrict__ out, float* __restrict__ Hg, unsigned* __restrict__ Bar)
{
    extern __shared__ float smem[];
    float* Wl = smem;                 // 64*SW
    float* Hs = Wl + 64 * SW;         // 32*SW
    float* Zs = Hs + 32 * SW;         // 32*SZ
    float* Cs = Zs + 32 * SZ;         // 32*16

    const int wg   = blockIdx.x;      // 0..31
    const int tid  = threadIdx.x;
    const int lane = tid & 31;
    const int wv   = tid >> 5;
    const int l16  = lane & 15;
    const int hi   = lane >> 4;
    const int mt   = wv & 1;          // batch half
    const int nt   = wv >> 1;         // 0..3 (16-col sub-tile)
    const int m0   = mt * 16;
    const int n0   = nt * 16;

    // --- prologue: resident weights (transposed, b128 global loads) --------
    {
        int c4 = (tid & 15) * 4;      // 0..60
        int kr = tid >> 4;            // 0..15
        for (int k = kr; k < L_; k += 16) {
            float4 w4 = *(const float4*)&Wperm[(size_t)(D_ + k) * N_ + wg * 64 + c4];
            Wl[(c4 + 0) * SW + k] = w4.x;
            Wl[(c4 + 1) * SW + k] = w4.y;
            Wl[(c4 + 2) * SW + k] = w4.z;
            Wl[(c4 + 3) * SW + k] = w4.w;
        }
    }
    for (int i = tid; i < B_ * 16; i += 256) {
        int m = i >> 4, jj = i & 15;
        Cs[m * 16 + jj] = c0[(size_t)m * L_ + wg * 16 + jj];
    }
    for (int i = tid; i < (B_ * L_) / 4; i += 256) {
        int g = i * 4;
        int m = g >> 9, k = g & (L_ - 1);
        *(float4*)&Hs[m * SW + k] = *(const float4*)&h0[g];
    }
    __syncthreads();

    const size_t cell_base = (size_t)B_ * T_ * L_;

    for (int t = 0; t < T_; ++t) {
        // seed accumulator with precomputed x-projection + bias
        const float* yb = Y + (size_t)t * B_ * N_ + wg * 64;
        v8f acc;
        #pragma unroll
        for (int r = 0; r < 8; ++r)
            acc[r] = yb[(size_t)(m0 + r + 8 * hi) * N_ + n0 + l16];

        // prefetch next step's Y slice now: it has GEMM + team barrier to land
        if (t + 1 < T_ && tid < 64) {
            const float* ynext = yb + (size_t)B_ * N_;
            int rr = tid >> 1, oo = (tid & 1) * 32;
            __builtin_prefetch(ynext + (size_t)rr * N_ + oo, 0, 0);
        }

        // recurrent GEMM: acc += H[32x512] @ Wslice[512x64]
        #pragma unroll 4
        for (int k0 = 0; k0 < L_; k0 += 4) {
            int kb = k0 + 2 * hi;
            v2f a = *(const v2f*)&Hs[(m0 + l16) * SW + kb];
            v2f b = *(const v2f*)&Wl[(n0 + l16) * SW + kb];
            acc = __builtin_amdgcn_wmma_f32_16x16x4_f32(
                false, a, false, b, (short)0, acc, false, false);
        }

        // exchange Z through LDS so every thread sees all 4 gates of its j
        #pragma unroll
        for (int r = 0; r < 8; ++r)
            Zs[(m0 + r + 8 * hi) * SZ + n0 + l16] = acc[r];
        __syncthreads();

        // gates: 32 batches x 16 j-cols = 512 values, 2 per thread
        #pragma unroll
        for (int p = 0; p < 2; ++p) {
            int idx = tid + p * 256;
            int m = idx >> 4, jj = idx & 15;
            float zi = Zs[m * SZ + jj];
            float zf = Zs[m * SZ + 16 + jj];
            float zo = Zs[m * SZ + 32 + jj];
            float zm = Zs[m * SZ + 48 + jj];
            float ig = 1.0f / (1.0f + expf(-zi));
            float fg = 1.0f / (1.0f + expf(-zf));
            float og = 1.0f / (1.0f + expf(-zo));
            float cn = fg * Cs[m * 16 + jj] + ig * tanhf(zm);
            float hn = og * tanhf(cn);
            Cs[m * 16 + jj] = cn;
            size_t oi = ((size_t)m * T_ + t) * L_ + wg * 16 + jj;
            out[oi]             = hn;
            out[cell_base + oi] = cn;
            Hg[((t + 1) & 1) * (B_ * L_) + (size_t)m * L_ + wg * 16 + jj] = hn;
        }

        __threadfence();              // release h_new to device scope
        __syncthreads();              // all lanes fenced before arrive
        if (tid == 0) {
            __hip_atomic_fetch_add(&Bar[t], 1u, __ATOMIC_RELEASE,
                                   __HIP_MEMORY_SCOPE_AGENT);
            while (__hip_atomic_load(&Bar[t], __ATOMIC_ACQUIRE,
                                     __HIP_MEMORY_SCOPE_AGENT) < (unsigned)NWG)
                __builtin_amdgcn_s_sleep(2);
        }
        __syncthreads();

        if (t + 1 < T_) {
            // async broadcast reload of full H (64KB) straight into LDS,
            // tracked with ASYNCcnt (no VGPR round-trip)
            const float* hsrc = Hg + ((t + 1) & 1) * (B_ * L_);
            for (int i = tid; i < (B_ * L_) / 4; i += 256) {
                int g = i * 4;
                int m = g >> 9, k = g & (L_ - 1);
                async_copy_b128(&Hs[m * SW + k], hsrc + g);
            }
            wait_async0();            // own wave's copies done
            __syncthreads();          // all waves' copies visible
        }
    }
}

// ---------------------------------------------------------------------------
// Launcher. Workspace layout (floats):
//   Wperm : 1024*2048 = 2,097,152 | biasp : 2048 | Hg : 32,768
//   Bar   : 2048 (unsigned)       | Y : 65536*2048 = 134,217,728
// Total ~521 MiB of d_ws.
// ---------------------------------------------------------------------------
extern "C" void kernel_launch(void* const* d_in, const int* in_sizes, int n_in,
                              void* d_out, int out_size, void* d_ws, size_t ws_size,
                              hipStream_t stream)
{
    const float* x   = (const float*)d_in[0];
    const float* h0  = (const float*)d_in[1];
    const float* c0  = (const float*)d_in[2];
    const float* w_w = (const float*)d_in[3];
    const float* w_b = (const float*)d_in[4];
    const float* m_w = (const float*)d_in[5];
    const float* m_b = (const float*)d_in[6];
    float* out = (float*)d_out;

    float* ws      = (float*)d_ws;
    float* Wperm   = ws;
    float* biasp   = Wperm + (size_t)1024 * N_;
    float* Hg      = biasp + N_;
    unsigned* Bar  = (unsigned*)(Hg + 2 * B_ * L_);
    float* Y       = (float*)(Bar + T_);

    // Phase 0: weight permutation + barrier reset
    lstm_prep_kernel<<<(1024 * N_) / 256, 256, 0, stream>>>(
        w_w, w_b, m_w, m_b, Wperm, biasp, Bar);

    // Phase 1: x-projection GEMM (fills Y with pre-activations + bias)
    dim3 g1(N_ / 128, (B_ * T_) / 64);
    lstm_xproj_kernel<<<g1, 256, 0, stream>>>(x, Wperm, biasp, Y);

    // Phase 2: persistent recurrent team (LDS-resident weights)
    size_t smem2 = (size_t)(64 * SW + 32 * SW + 32 * SZ + 32 * 16) * sizeof(float);
    static bool attr_set = false;
    if (!attr_set) {
        hipFuncSetAttribute((const void*)lstm_rec_kernel,
                            hipFuncAttributeMaxDynamicSharedMemorySize,
                            (int)smem2);
        attr_set = true;
    }
    lstm_rec_kernel<<<NWG, 256, smem2, stream>>>(
        Wperm, Y, h0, c0, out, Hg, Bar);
}